// Gclip_Gsyn_56942676410883
// MI455X (gfx1250) — compile-verified
//
#include <hip/hip_runtime.h>
#include <hip/hip_bf16.h>

typedef float v2f __attribute__((ext_vector_type(2)));
typedef float v8f __attribute__((ext_vector_type(8)));

#define B_      64
#define C_      64
#define T_      6000
#define EP_     10
#define L_      600      // T_/EP_ exactly
#define CHUNK   120
#define NCHUNK  5        // 5*120 = 600
#define LDST    124      // LDS chunk row stride: bank-conflict-free fragment reads, 16B-aligned stores
#define COVST   68       // cov row stride: half-wave row groups land on disjoint bank sets

__launch_bounds__(128, 1)
__global__ void corr_gram_wmma_f32(const float* __restrict__ x,
                                   float* __restrict__ out) {
    __shared__ float sChunk[C_ * LDST];    // 31,744 B staging tile
    __shared__ float sCov[C_ * COVST];     // 17,408 B raw Gram matrix
    __shared__ float sPart[128];
    __shared__ float sSum[C_];
    __shared__ float sRd[C_];

    const int be   = blockIdx.x;          // 0..639
    const int b    = be / EP_;
    const int e    = be - b * EP_;
    const int tid  = threadIdx.x;
    const int lane = tid & 31;
    const int w    = tid >> 5;            // wave id 0..3 -> row-tile stripe

    const float* __restrict__ xin = x + (size_t)b * C_ * T_ + (size_t)e * L_;

    // fp32 WMMA 16x16x4 fragment addressing (per CDNA5 ISA layouts):
    // A 16x4: lanes 0-15 hold (M=lane, K=0/1), lanes 16-31 hold (M=lane-16, K=2/3)
    // B 4x16 is the transposed tile of the same matrix -> identical addressing form.
    const int m16  = lane & 15;
    const int koff = (lane >> 4) * 2;
    const int aBase  = (w * 16 + m16) * LDST + koff;
    const int bBase0 = ( 0 + m16) * LDST + koff;
    const int bBase1 = (16 + m16) * LDST + koff;
    const int bBase2 = (32 + m16) * LDST + koff;
    const int bBase3 = (48 + m16) * LDST + koff;

    const int sumCh   = tid >> 1;         // 2 threads per channel
    const int sumHalf = tid & 1;

    v8f acc0 = {}, acc1 = {}, acc2 = {}, acc3 = {};
    float psum = 0.0f;

    for (int ck = 0; ck < NCHUNK; ++ck) {
        // ---- stage 64x120 fp32 chunk into LDS (aligned float4 both sides) ----
        const float* __restrict__ src = xin + ck * CHUNK;
#pragma unroll
        for (int i = 0; i < 15; ++i) {                 // 1920 float4 / 128 threads
            int fidx = tid + i * 128;
            int row  = fidx / 30;                      // 30 float4 per row
            int c4   = fidx - row * 30;
            float4 v = *(const float4*)(src + (size_t)row * T_ + c4 * 4);
            *(float4*)&sChunk[row * LDST + c4 * 4] = v;
        }
        __syncthreads();

        // ---- prefetch next chunk into cache while WMMA works on this one ----
        // (single resident wave per SIMD -> no other wave hides HBM latency;
        //  global_prefetch_b8 overlaps the next 30KB fetch with compute)
        if (ck + 1 < NCHUNK) {
            const float* nxt = xin + (ck + 1) * CHUNK;
#pragma unroll
            for (int i = 0; i < 2; ++i) {
                int fidx = tid + i * 128;              // 256 cacheline slots
                int row  = fidx >> 2;                  // 4 x 128B covers a 480B row
                int seg  = fidx & 3;
                __builtin_prefetch(nxt + (size_t)row * T_ + seg * 32, 0, 3);
            }
        }

        // ---- per-channel partial sums (for mean correction) ----
        {
            const float* r = &sChunk[sumCh * LDST + sumHalf * (CHUNK / 2)];
            float s = 0.0f;
#pragma unroll 4
            for (int j = 0; j < CHUNK / 2; ++j) s += r[j];
            psum += s;
        }

        // ---- Gram accumulation: 30 k-steps, 4 WMMAs each (A reused x4) ----
        for (int k = 0; k < CHUNK; k += 4) {
            v2f a  = *(const v2f*)&sChunk[aBase  + k];
            v2f b0 = *(const v2f*)&sChunk[bBase0 + k];
            v2f b1 = *(const v2f*)&sChunk[bBase1 + k];
            v2f b2 = *(const v2f*)&sChunk[bBase2 + k];
            v2f b3 = *(const v2f*)&sChunk[bBase3 + k];
            acc0 = __builtin_amdgcn_wmma_f32_16x16x4_f32(false, a, false, b0,
                                                         (short)0, acc0, false, false);
            acc1 = __builtin_amdgcn_wmma_f32_16x16x4_f32(false, a, false, b1,
                                                         (short)0, acc1, false, false);
            acc2 = __builtin_amdgcn_wmma_f32_16x16x4_f32(false, a, false, b2,
                                                         (short)0, acc2, false, false);
            acc3 = __builtin_amdgcn_wmma_f32_16x16x4_f32(false, a, false, b3,
                                                         (short)0, acc3, false, false);
        }
        __syncthreads();
    }

    // ---- spill Gram tiles to LDS (C/D layout: VGPR r -> M=r (+8 for upper half-wave)) ----
    {
        int rbase = w * 16 + (lane >> 4) * 8;
#pragma unroll
        for (int r = 0; r < 8; ++r) {
            int row = rbase + r;
            sCov[row * COVST +  0 + m16] = acc0[r];
            sCov[row * COVST + 16 + m16] = acc1[r];
            sCov[row * COVST + 32 + m16] = acc2[r];
            sCov[row * COVST + 48 + m16] = acc3[r];
        }
    }
    sPart[tid] = psum;
    __syncthreads();

    if (tid < C_) {
        float s = sPart[2 * tid] + sPart[2 * tid + 1];
        sSum[tid] = s;
        float diag = sCov[tid * COVST + tid] - s * s * (1.0f / L_);
        sRd[tid] = rsqrtf(fmaxf(diag, 1e-30f));
    }
    __syncthreads();

    // ---- normalize, clip, write out[b][e][i][j] as aligned float4 ----
    float* __restrict__ dst = out + (size_t)be * (C_ * C_);
    const float invT = 1.0f / L_;
#pragma unroll
    for (int g = 0; g < 8; ++g) {
        int fidx = tid + g * 128;          // 0..1023
        int row  = fidx >> 4;
        int c0   = (fidx & 15) * 4;
        float si = sSum[row], ri = sRd[row];
        float4 gv = *(const float4*)&sCov[row * COVST + c0];
        float4 o;
        o.x = fminf(fmaxf((gv.x - si * sSum[c0 + 0] * invT) * ri * sRd[c0 + 0], -1.0f), 1.0f);
        o.y = fminf(fmaxf((gv.y - si * sSum[c0 + 1] * invT) * ri * sRd[c0 + 1], -1.0f), 1.0f);
        o.z = fminf(fmaxf((gv.z - si * sSum[c0 + 2] * invT) * ri * sRd[c0 + 2], -1.0f), 1.0f);
        o.w = fminf(fmaxf((gv.w - si * sSum[c0 + 3] * invT) * ri * sRd[c0 + 3], -1.0f), 1.0f);
        *(float4*)(dst + row * C_ + c0) = o;
    }
}

extern "C" void kernel_launch(void* const* d_in, const int* in_sizes, int n_in,
                              void* d_out, int out_size, void* d_ws, size_t ws_size,
                              hipStream_t stream) {
    (void)in_sizes; (void)n_in; (void)out_size; (void)d_ws; (void)ws_size;
    const float* x = (const float*)d_in[0];
    float* out = (float*)d_out;
    dim3 grid(B_ * EP_);   // 640 (b,e) blocks
    dim3 block(128);       // 4 wave32s
    corr_gram_wmma_f32<<<grid, block, 0, stream>>>(x, out);
}